// OpenVocabHungarianMatcher_36000415875396
// MI455X (gfx1250) — compile-verified
//
#include <hip/hip_runtime.h>
#include <cstdint>

// Problem constants (match reference).
#define F_ALPHA 0.25f
#define F_EPS   1e-8f
#define W_CLASS 2.0f
#define W_BBOX  5.0f
#define W_GIOU  2.0f

constexpr int B = 8, Q = 900, K = 1203, N = 100;
constexpr int QPB    = 8;                      // queries per block: 1 per wave32, 8 waves
constexpr int QTILES = (Q + QPB - 1) / QPB;    // 113

// ---- CDNA5 async global->LDS DMA helpers (tracked by ASYNCcnt) ----
__device__ __forceinline__ uint32_t lds_offset(const void* p) {
    // generic -> LDS(AS3) addrspacecast, then ptrtoint (AS3 pointers are 32-bit)
    return (uint32_t)(uintptr_t)(__attribute__((address_space(3))) const void*)p;
}
__device__ __forceinline__ void async_load_b32(uint32_t lds_off, const void* g) {
    asm volatile("global_load_async_to_lds_b32 %0, %1, off"
                 :: "v"(lds_off), "v"((uint64_t)(uintptr_t)g) : "memory");
}
__device__ __forceinline__ void async_load_b128(uint32_t lds_off, const void* g) {
    asm volatile("global_load_async_to_lds_b128 %0, %1, off"
                 :: "v"(lds_off), "v"((uint64_t)(uintptr_t)g) : "memory");
}
__device__ __forceinline__ void wait_asynccnt0() {
    asm volatile("s_wait_asynccnt 0" ::: "memory");
}

__global__ __launch_bounds__(256)
void matcher_cost_kernel(const float* __restrict__ logits,   // (B,Q,K)
                         const float* __restrict__ pboxes,   // (B,Q,4) cxcywh
                         const int*   __restrict__ tlabels,  // (B,N)
                         const float* __restrict__ tboxes,   // (B,N,4) cxcywh
                         float* __restrict__ out)            // (B,Q,N)
{
    __shared__ float s_raw[N * 4];                 // raw cxcywh targets (for L1)
    __shared__ int   s_lbl[N];
    __shared__ float s_x0[N], s_y0[N], s_x1[N], s_y1[N], s_ar[N];

    const int b     = blockIdx.x / QTILES;
    const int qtile = blockIdx.x % QTILES;
    const int tid   = threadIdx.x;
    const int wave  = tid >> 5;
    const int lane  = tid & 31;
    const int q     = qtile * QPB + wave;

    // Stage this batch's targets into LDS with the async DMA path.
    if (tid < N) {
        async_load_b128(lds_offset(&s_raw[tid * 4]),
                        tboxes + (size_t)(b * N + tid) * 4);
        async_load_b32 (lds_offset(&s_lbl[tid]),
                        tlabels + b * N + tid);
    }

    // Warm L0/L2 with this wave's 4812B logits row while the DMA is in flight.
    if (q < Q) {
        const char* row = (const char*)(logits + (size_t)(b * Q + q) * K);
        int off = lane * 128;
        if (off < K * 4) __builtin_prefetch(row + off, 0, 0);
        off += 32 * 128;
        if (off < K * 4) __builtin_prefetch(row + off, 0, 0);
    }

    wait_asynccnt0();       // issuing waves drain ASYNCcnt before signaling
    __syncthreads();

    // Precompute xyxy + area (exactly as reference: area from converted coords).
    if (tid < N) {
        const float cx = s_raw[tid * 4 + 0], cy = s_raw[tid * 4 + 1];
        const float w  = s_raw[tid * 4 + 2], h  = s_raw[tid * 4 + 3];
        const float x0 = cx - 0.5f * w, y0 = cy - 0.5f * h;
        const float x1 = cx + 0.5f * w, y1 = cy + 0.5f * h;
        s_x0[tid] = x0; s_y0[tid] = y0; s_x1[tid] = x1; s_y1[tid] = y1;
        s_ar[tid] = (x1 - x0) * (y1 - y0);
    }
    __syncthreads();

    if (q >= Q) return;

    // All lanes of the wave share one query: broadcast 16B load.
    const float4 pb = *reinterpret_cast<const float4*>(pboxes + (size_t)(b * Q + q) * 4);
    const float px0 = pb.x - 0.5f * pb.z, py0 = pb.y - 0.5f * pb.w;
    const float px1 = pb.x + 0.5f * pb.z, py1 = pb.y + 0.5f * pb.w;
    const float parea = (px1 - px0) * (py1 - py0);

    const float* lrow = logits + (size_t)(b * Q + q) * K;
    float*       orow = out    + (size_t)(b * Q + q) * N;

    #pragma unroll
    for (int n0 = 0; n0 < N; n0 += 32) {
        const int n = n0 + lane;
        if (n < N) {
            // ---- focal-style class cost at the gathered label ----
            const int   lbl = s_lbl[n];
            const float x   = lrow[lbl];
            const float p   = 1.0f / (1.0f + expf(-x));
            const float omp = 1.0f - p;
            const float pos = -F_ALPHA * omp * omp * logf(fmaxf(p,   F_EPS));
            const float neg = -(1.0f - F_ALPHA) * p * p * logf(fmaxf(omp, F_EPS));
            const float cls = pos - neg;

            // ---- L1 on raw cxcywh ----
            const float l1 = fabsf(pb.x - s_raw[n * 4 + 0])
                           + fabsf(pb.y - s_raw[n * 4 + 1])
                           + fabsf(pb.z - s_raw[n * 4 + 2])
                           + fabsf(pb.w - s_raw[n * 4 + 3]);

            // ---- GIoU on xyxy ----
            const float tx0 = s_x0[n], ty0 = s_y0[n];
            const float tx1 = s_x1[n], ty1 = s_y1[n];
            const float iw = fmaxf(fminf(px1, tx1) - fmaxf(px0, tx0), 0.0f);
            const float ih = fmaxf(fminf(py1, ty1) - fmaxf(py0, ty0), 0.0f);
            const float inter = iw * ih;
            const float uni   = parea + s_ar[n] - inter;
            const float iou   = inter / uni;
            const float cw = fmaxf(fmaxf(px1, tx1) - fminf(px0, tx0), 0.0f);
            const float ch = fmaxf(fmaxf(py1, ty1) - fminf(py0, ty0), 0.0f);
            const float ac = cw * ch;
            const float giou = iou - (ac - uni) / ac;

            orow[n] = W_CLASS * cls + W_BBOX * l1 - W_GIOU * giou;
        }
    }
}

extern "C" void kernel_launch(void* const* d_in, const int* in_sizes, int n_in,
                              void* d_out, int out_size, void* d_ws, size_t ws_size,
                              hipStream_t stream) {
    const float* logits  = (const float*)d_in[0];  // pred_logits (B,Q,K) f32
    const float* pboxes  = (const float*)d_in[1];  // pred_boxes  (B,Q,4) f32
    const int*   tlabels = (const int*)  d_in[2];  // tgt_labels  (B,N)   int
    const float* tboxes  = (const float*)d_in[3];  // tgt_boxes   (B,N,4) f32
    float* out = (float*)d_out;                    // (B,Q,N) f32

    dim3 grid(B * QTILES);   // 904 blocks, 8 wave32 each
    dim3 block(256);
    matcher_cost_kernel<<<grid, block, 0, stream>>>(logits, pboxes, tlabels, tboxes, out);
}